// VAE_22780506538221
// MI455X (gfx1250) — compile-verified
//
#include <hip/hip_runtime.h>
#include <math.h>

// Problem constants (match reference)
#define B_  128
#define N_  128
#define L_  64
#define P_  30
#define NN_ (N_ * N_)   // 16384

typedef float v2f __attribute__((ext_vector_type(2)));
typedef float v8f __attribute__((ext_vector_type(8)));

// ---------------------------------------------------------------------------
// Kernel 1: z = mu + eps*exp(logstd); KL[b] = sum_l(0.5 z^2 - 0.5 eps^2 - logstd)
// (the -0.5*LOG2PI terms cancel between log_q and log_p)
// ---------------------------------------------------------------------------
__global__ void vae_zkl_kernel(const float* __restrict__ mu,
                               const float* __restrict__ logstd,
                               const float* __restrict__ eps,
                               float* __restrict__ zws,
                               float* __restrict__ klws) {
    const int b = blockIdx.x;
    const int l = threadIdx.x;               // 0..63
    const int idx = b * L_ + l;
    const float e = eps[idx];
    const float s = logstd[idx];
    const float z = mu[idx] + e * expf(s);
    zws[idx] = z;
    float t = 0.5f * z * z - 0.5f * e * e - s;

    __shared__ float red[L_];
    red[l] = t;
    __syncthreads();
    for (int off = L_ / 2; off > 0; off >>= 1) {
        if (l < off) red[l] += red[l + off];
        __syncthreads();
    }
    if (l == 0) klws[b] = red[0];
}

// ---------------------------------------------------------------------------
// Kernel 2: logits[0:32, 0:16384] = z[0:32] @ W_dec  via V_WMMA_F32_16X16X4_F32
// One wave (32 threads) per 32x16 output tile; K=64 -> 16 WMMA steps.
// Epilogue also stores lsneg = log_sigmoid(-x) = -softplus(x).
// A-operand layout (32-bit 16x4): lane holds M = lane&15; lane[4] selects K pair
//   {0,1} vs {2,3} (v0 = first K of pair, v1 = second).
// B-operand layout (4x16) is symmetric: lane holds N = lane&15, same K pairing.
// C/D layout: VGPR v, lanes 0-15 -> (M=v, N=lane); lanes 16-31 -> (M=v+8).
// ---------------------------------------------------------------------------
__global__ void vae_gemm_wmma_kernel(const float* __restrict__ zws,
                                     const float* __restrict__ W,
                                     float* __restrict__ logitsWS,
                                     float* __restrict__ lsnegWS) {
    const int lane  = threadIdx.x;           // 0..31, EXEC all ones (no divergence)
    const int nc    = lane & 15;             // M index for A, N index for B
    const int khalf = (lane >> 4) << 1;      // 0 or 2
    const int n0    = blockIdx.x * 16;       // output-column tile base

    v8f c0 = {};                             // rows 0..15
    v8f c1 = {};                             // rows 16..31

    for (int kk = 0; kk < 16; ++kk) {
        const int kb = (kk << 2) + khalf;    // this lane's K-pair base
        v2f a0, a1, bb;
        a0.x = zws[nc * L_ + kb];
        a0.y = zws[nc * L_ + kb + 1];
        a1.x = zws[(nc + 16) * L_ + kb];
        a1.y = zws[(nc + 16) * L_ + kb + 1];
        const int n = n0 + nc;
        bb.x = W[(size_t)kb * NN_ + n];
        bb.y = W[(size_t)(kb + 1) * NN_ + n];
        if (kk < 15) {
            // prefetch next K-slab of W_dec (emits global_prefetch_b8)
            __builtin_prefetch(&W[(size_t)(kb + 4) * NN_ + n], 0, 3);
        }
        // 8 args: (neg_a, A, neg_b, B, c_mod, C, reuse_a, reuse_b)
        c0 = __builtin_amdgcn_wmma_f32_16x16x4_f32(false, a0, false, bb,
                                                   (short)0, c0, false, false);
        c1 = __builtin_amdgcn_wmma_f32_16x16x4_f32(false, a1, false, bb,
                                                   (short)0, c1, false, false);
    }

    const int n     = n0 + nc;
    const int mbase = (lane >> 4) << 3;      // 0 or 8
#pragma unroll
    for (int v = 0; v < 8; ++v) {
        const int m = mbase + v;
        const float x0 = c0[v];
        logitsWS[(size_t)m * NN_ + n] = x0;
        lsnegWS[(size_t)m * NN_ + n]  = -(fmaxf(x0, 0.0f) + log1pf(expf(-fabsf(x0))));
        const float x1 = c1[v];
        logitsWS[(size_t)(m + 16) * NN_ + n] = x1;
        lsnegWS[(size_t)(m + 16) * NN_ + n]  = -(fmaxf(x1, 0.0f) + log1pf(expf(-fabsf(x1))));
    }
}

// ---------------------------------------------------------------------------
// Kernel 3: rec[b,j] = sum_{i,k} m[j,i,k] * ( lsneg[j, p_i, p_k] +
//                                             A[j,i,k] * logit[j, p_i, p_k] )
// using logsig(x) - logsig(-x) = x  (no transcendentals in the hot loop).
// grid = (P_, B_), block = N_ threads; thread index = column k.
// ---------------------------------------------------------------------------
__global__ void vae_rec_kernel(const int* __restrict__ Adj,
                               const int* __restrict__ masks,
                               const int* __restrict__ perms,
                               const float* __restrict__ logitsWS,
                               const float* __restrict__ lsnegWS,
                               float* __restrict__ recWS) {
    const int j = blockIdx.x;                // 0..29
    const int b = blockIdx.y;                // 0..127
    const int k = threadIdx.x;               // 0..127

    const int* __restrict__ pbj = perms + ((size_t)b * P_ + j) * N_;
    const int pk = pbj[k];

    const int*   __restrict__ Aj = Adj      + (size_t)j * NN_;
    const int*   __restrict__ Mj = masks    + (size_t)j * NN_;
    const float* __restrict__ Lj = logitsWS + (size_t)j * NN_;
    const float* __restrict__ Sj = lsnegWS  + (size_t)j * NN_;

    float acc = 0.0f;
    for (int i = 0; i < N_; ++i) {
        const int   mv = Mj[i * N_ + k];     // coalesced across lanes
        const int   pi = pbj[i];             // uniform broadcast
        const int   a  = Aj[i * N_ + k];     // coalesced across lanes
        const float x  = Lj[pi * N_ + pk];   // gather within a 512B cached row
        const float s  = Sj[pi * N_ + pk];
        const float v  = s + (a ? x : 0.0f);
        acc += mv ? v : 0.0f;
    }

    __shared__ float red[N_];
    red[k] = acc;
    __syncthreads();
    for (int off = N_ / 2; off > 0; off >>= 1) {
        if (k < off) red[k] += red[k + off];
        __syncthreads();
    }
    if (k == 0) recWS[(size_t)b * P_ + j] = red[0];
}

// ---------------------------------------------------------------------------
// Kernel 4: out = mean(KL) - sum_b max_j rec[b,j]   ( = -elbo )
// ---------------------------------------------------------------------------
__global__ void vae_final_kernel(const float* __restrict__ klws,
                                 const float* __restrict__ recWS,
                                 float* __restrict__ out) {
    const int t = threadIdx.x;               // 0..127, one per batch b
    float mx = -3.402823466e38f;
    for (int jj = 0; jj < P_; ++jj) mx = fmaxf(mx, recWS[(size_t)t * P_ + jj]);
    float v = klws[t] * (1.0f / (float)B_) - mx;

    __shared__ float red[B_];
    red[t] = v;
    __syncthreads();
    for (int off = B_ / 2; off > 0; off >>= 1) {
        if (t < off) red[t] += red[t + off];
        __syncthreads();
    }
    if (t == 0) out[0] = red[0];
}

// ---------------------------------------------------------------------------
// Host launcher
// ---------------------------------------------------------------------------
extern "C" void kernel_launch(void* const* d_in, const int* in_sizes, int n_in,
                              void* d_out, int out_size, void* d_ws, size_t ws_size,
                              hipStream_t stream) {
    const float* mu     = (const float*)d_in[0];   // (B, L)
    const float* logstd = (const float*)d_in[1];   // (B, L)
    const float* eps    = (const float*)d_in[2];   // (B, L)
    const float* W_dec  = (const float*)d_in[3];   // (L, N*N)
    const int*   Adj    = (const int*)d_in[4];     // (B, N, N)
    const int*   masks  = (const int*)d_in[5];     // (B, N, N)
    const int*   perms  = (const int*)d_in[6];     // (B, P, N)
    float* out = (float*)d_out;

    // workspace layout (floats)
    float* ws       = (float*)d_ws;
    float* zws      = ws;                                  // B*L          =   8192
    float* logitsWS = zws + (size_t)B_ * L_;               // 32*NN        = 524288
    float* lsnegWS  = logitsWS + (size_t)32 * NN_;         // 32*NN        = 524288
    float* klws     = lsnegWS + (size_t)32 * NN_;          // B            =    128
    float* recWS    = klws + B_;                           // B*P          =   3840

    vae_zkl_kernel<<<B_, L_, 0, stream>>>(mu, logstd, eps, zws, klws);
    vae_gemm_wmma_kernel<<<NN_ / 16, 32, 0, stream>>>(zws, W_dec, logitsWS, lsnegWS);
    vae_rec_kernel<<<dim3(P_, B_), N_, 0, stream>>>(Adj, masks, perms,
                                                    logitsWS, lsnegWS, recWS);
    vae_final_kernel<<<1, B_, 0, stream>>>(klws, recWS, out);
}